// ResidualVectorQuantizer_67276367725221
// MI455X (gfx1250) — compile-verified
//
#include <hip/hip_runtime.h>

// Residual Vector Quantizer for MI455X (gfx1250, wave32).
// Distance GEMM runs on V_WMMA_F32_16X16X4_F32 (exact f32, matches reference
// argmin behavior). Residual/quantized state lives in LDS for all 8 levels.

#define NQ            8
#define N_ROWS        8192
#define DIM           128
#define BINS          256
#define ROWS_PER_BLK  16
#define THREADS       256
#define RES_STRIDE    (DIM + 4)   // +4 floats: conflict-free WMMA A-fragment loads, keeps 16B align

typedef __attribute__((ext_vector_type(2))) float v2f;
typedef __attribute__((ext_vector_type(8))) float v8f;

// ---- Precompute ||c||^2 for every (level, bin) into workspace -------------
__global__ __launch_bounds__(THREADS)
void rvq_norms_kernel(const float* __restrict__ cb, float* __restrict__ norms) {
    int idx = blockIdx.x * blockDim.x + threadIdx.x;     // 0 .. NQ*BINS-1
    const float* row = cb + (size_t)idx * DIM;
    float s = 0.f;
    #pragma unroll 8
    for (int d = 0; d < DIM; ++d) { float v = row[d]; s += v * v; }
    norms[idx] = s;
}

// ---- Main RVQ kernel: one block = 16 rows, all 8 levels -------------------
__global__ __launch_bounds__(THREADS)
void rvq_main_kernel(const float* __restrict__ hs,
                     const float* __restrict__ cb,
                     const float* __restrict__ norms,
                     int*   __restrict__ codes_out,
                     float* __restrict__ quant_out) {
    __shared__ float res  [ROWS_PER_BLK][RES_STRIDE];
    __shared__ float quant[ROWS_PER_BLK][RES_STRIDE];
    __shared__ float scores[ROWS_PER_BLK][BINS];
    __shared__ float pmin [ROWS_PER_BLK][16];
    __shared__ int   pidx [ROWS_PER_BLK][16];
    __shared__ int   bestIdx[ROWS_PER_BLK];

    const int tid  = threadIdx.x;
    const int lane = tid & 31;
    const int wave = tid >> 5;
    const int rowbase = blockIdx.x * ROWS_PER_BLK;

    // Load residual rows, zero the quantized accumulator (16 rows x 128 dims,
    // 8 floats per thread, float4 vectorized).
    {
        const int row = tid >> 4;
        const int d0  = (tid & 15) * 8;
        const float4* src = (const float4*)(hs + (size_t)(rowbase + row) * DIM + d0);
        *(float4*)&res[row][d0]     = src[0];
        *(float4*)&res[row][d0 + 4] = src[1];
        float4 z = make_float4(0.f, 0.f, 0.f, 0.f);
        *(float4*)&quant[row][d0]     = z;
        *(float4*)&quant[row][d0 + 4] = z;
    }
    __syncthreads();

    // WMMA f32 16x16x4 fragment coordinates for this lane:
    //   A (16x4): lanes 0-15 hold row r, K={0,1}; lanes 16-31 hold row r, K={2,3}
    //   B (4x16): same split with N instead of M.
    const int r  = lane & 15;
    const int kh = lane >> 4;

    for (int level = 0; level < NQ; ++level) {
        const float* cbl = cb + (size_t)level * BINS * DIM;

        // Prefetch next level's codebook (128KB spread over 256 threads).
        if (level + 1 < NQ) {
            const char* p = (const char*)(cbl + BINS * DIM) + tid * 512;
            __builtin_prefetch(p,       0, 1);
            __builtin_prefetch(p + 128, 0, 1);
            __builtin_prefetch(p + 256, 0, 1);
            __builtin_prefetch(p + 384, 0, 1);
        }

        // --- distance GEMM: each wave owns two 16-wide N tiles -------------
        #pragma unroll
        for (int t = 0; t < 2; ++t) {
            const int nbase = wave * 32 + t * 16;
            const int n = nbase + r;                    // this lane's column
            const float* bcol = cbl + (size_t)n * DIM;  // B[k][n] = codebook[n][k]
            v8f acc = {};
            #pragma unroll
            for (int kk = 0; kk < 32; ++kk) {           // K = 128 in steps of 4
                const int kb = kk * 4 + kh * 2;
                v2f a = *(const v2f*)&res[r][kb];        // LDS, conflict-free
                v2f b = *(const v2f*)&bcol[kb];          // L2-resident codebook
                acc = __builtin_amdgcn_wmma_f32_16x16x4_f32(
                        /*neg_a=*/false, a, /*neg_b=*/false, b,
                        /*c_mod=*/(short)0, acc, /*reuse_a=*/false, /*reuse_b=*/false);
            }
            const float cn = norms[level * BINS + n];
            #pragma unroll
            for (int j = 0; j < 8; ++j) {
                const int m = j + kh * 8;               // C/D layout: vgpr j -> row j / j+8
                scores[m][n] = cn - 2.0f * acc[j];      // ||c||^2 - 2 r.c (||r||^2 dropped)
            }
        }
        __syncthreads();

        // --- argmin phase 1: 16 threads/row, each scans 16 codes ascending --
        {
            const int row = tid >> 4;
            const int seg = tid & 15;
            const int base = seg * 16;
            float best = scores[row][base];
            int   bi   = base;
            #pragma unroll
            for (int i = 1; i < 16; ++i) {
                float v = scores[row][base + i];
                if (v < best) { best = v; bi = base + i; }   // strict <: first-index ties
            }
            pmin[row][seg] = best;
            pidx[row][seg] = bi;
        }
        __syncthreads();

        // --- argmin phase 2 + emit code -------------------------------------
        if (tid < ROWS_PER_BLK) {
            float best = pmin[tid][0];
            int   bi   = pidx[tid][0];
            #pragma unroll
            for (int s = 1; s < 16; ++s) {
                float v = pmin[tid][s];
                if (v < best) { best = v; bi = pidx[tid][s]; }
            }
            bestIdx[tid] = bi;
            codes_out[(size_t)level * N_ROWS + rowbase + tid] = bi;
        }
        __syncthreads();

        // --- update residual & quantized with chosen code -------------------
        {
            const int row = tid >> 4;
            const int d0  = (tid & 15) * 8;
            const float* cptr = cbl + (size_t)bestIdx[row] * DIM + d0;
            #pragma unroll
            for (int d = 0; d < 8; ++d) {
                float v = cptr[d];
                res[row][d0 + d]   -= v;
                quant[row][d0 + d] += v;
            }
        }
        __syncthreads();
    }

    // Write quantized output.
    {
        const int row = tid >> 4;
        const int d0  = (tid & 15) * 8;
        float* dst = quant_out + (size_t)(rowbase + row) * DIM + d0;
        *(float4*)(dst)     = *(const float4*)&quant[row][d0];
        *(float4*)(dst + 4) = *(const float4*)&quant[row][d0 + 4];
    }
}

extern "C" void kernel_launch(void* const* d_in, const int* in_sizes, int n_in,
                              void* d_out, int out_size, void* d_ws, size_t ws_size,
                              hipStream_t stream) {
    const float* hs = (const float*)d_in[0];   // [8192, 128]
    const float* cb = (const float*)d_in[1];   // [8, 256, 128]
    float* norms = (float*)d_ws;               // NQ*BINS floats (8 KB)

    // d_out layout: codes [NQ, N] (int32 bits) followed by quantized [N, DIM] f32.
    int*   codes_out = (int*)d_out;
    float* quant_out = (float*)d_out + (size_t)NQ * N_ROWS;

    rvq_norms_kernel<<<(NQ * BINS) / THREADS, THREADS, 0, stream>>>(cb, norms);
    rvq_main_kernel<<<N_ROWS / ROWS_PER_BLK, THREADS, 0, stream>>>(
        hs, cb, norms, codes_out, quant_out);
}